// LightMLA_68307159876084
// MI455X (gfx1250) — compile-verified
//
#include <hip/hip_runtime.h>
#include <hip/hip_bf16.h>

// ---------------------------------------------------------------------------
// LightMLA pipeline for gfx1250 (MI455X), fp32 end-to-end.
//   qkv : GEMM [768x256]x[256xN]  -> WMMA f32 16x16x4, TDM-staged X in LDS
//   dw  : depthwise 5x5 (pad 2)   -> VALU
//   pw  : grouped 1x1 (96 groups) -> VALU, in-place
//   attn: kv = relu(k)^T [v|1] ; out = relu(q) kv ; normalize
//   proj: GEMM [256x512]x[512xN] + BN affine -> WMMA f32 16x16x4 + TDM
// ---------------------------------------------------------------------------

typedef float v2f __attribute__((ext_vector_type(2)));
typedef float v8f __attribute__((ext_vector_type(8)));
typedef unsigned int u32x4 __attribute__((ext_vector_type(4)));
typedef int i32x4 __attribute__((ext_vector_type(4)));
typedef int i32x8 __attribute__((ext_vector_type(8)));

#define HW 16384
#define HDIM 128
#define KC 64                    // K-chunk staged per TDM transfer
#define LDS_ROW 34               // 32 data dwords + 2 pad dwords (TDM padding)
#define CHUNK_F (KC * LDS_ROW)   // floats per LDS buffer (2176)
#define LDS_BYTES (2 * CHUNK_F * 4)

#if defined(__has_builtin)
#if __has_builtin(__builtin_amdgcn_tensor_load_to_lds)
#define HAVE_TDM 1
#endif
#if __has_builtin(__builtin_amdgcn_s_wait_tensorcnt)
#define HAVE_TENSORCNT 1
#endif
#endif

// ---------------------------------------------------------------------------
// TDM: async-load a (KC x 32) fp32 tile (row stride HW dwords in memory) into
// LDS at byte offset lds_off, with 2 pad dwords inserted every 32 dwords
// (LDS row pitch = 34 floats).  D# built per CDNA5 ISA 8.3/8.4.
// Issued once per wave (EXEC-independent); caller gates to one wave.
// ---------------------------------------------------------------------------
__device__ __forceinline__ void tdm_load_chunk(const float* __restrict__ gsrc,
                                               unsigned lds_off_bytes,
                                               float* __restrict__ lds,
                                               int lane) {
#ifdef HAVE_TDM
  const unsigned long long ga = (unsigned long long)(uintptr_t)gsrc;
  u32x4 g0 = {0u, 0u, 0u, 0u};
  g0[0] = 1u;                                       // count=1, user descriptor
  g0[1] = lds_off_bytes;                            // lds_addr
  g0[2] = (unsigned)ga;                             // global_addr[31:0]
  g0[3] = (unsigned)((ga >> 32) & 0x01FFFFFFu) | (2u << 30);  // addr[56:32], type=2
  i32x8 g1 = {0, 0, 0, 0, 0, 0, 0, 0};
  // data_size=2 (4B), pad_enable, pad_interval=4 (32 dw), pad_amount=1 (2 dw)
  g1[0] = (int)((2u << 16) | (1u << 20) | (4u << 22) | (1u << 25));
  g1[1] = 0;                                        // tensor_dim0 = 0x40000000 (lo16=0)
  g1[2] = (int)(0x4000u);                           // dim0 hi16 ; tensor_dim1 lo16=0
  g1[3] = (int)(0x4000u | (32u << 16));             // dim1 hi16 ; tile_dim0 = 32
  g1[4] = (int)KC;                                  // tile_dim1 = KC ; tile_dim2 = 0
  g1[5] = (int)HW;                                  // tensor_dim0_stride = HW dwords
  i32x4 z4 = {0, 0, 0, 0};
#if __clang_major__ >= 23
  i32x8 z8 = {0, 0, 0, 0, 0, 0, 0, 0};
  __builtin_amdgcn_tensor_load_to_lds(g0, g1, z4, z4, z8, 0);
#else
  __builtin_amdgcn_tensor_load_to_lds(g0, g1, z4, z4, 0);
#endif
#else
  // Fallback: one wave copies the chunk through VGPRs.
  for (int i = lane; i < KC * 32; i += 32) {
    const int r = i >> 5, c = i & 31;
    lds[lds_off_bytes / 4 + r * LDS_ROW + c] = gsrc[(size_t)r * HW + c];
  }
#endif
}

__device__ __forceinline__ void tdm_wait0() {
#ifdef HAVE_TENSORCNT
  __builtin_amdgcn_s_wait_tensorcnt(0);
#else
  __builtin_amdgcn_s_waitcnt(0);
#endif
}

// ---------------------------------------------------------------------------
// 1x1-conv GEMM:  Out[b, m, p] = sum_k W[m, k] * X[b, k, p]
// Block = 256 threads = 8 waves.  Block tile = 128(M) x 32(N):
//   wave w -> M rows [blockIdx.y*128 + w*16, +16), both 16-wide N subtiles.
// X tile (KC x 32) double-buffered in LDS via TDM; A-frags straight from
// global (weights are tiny and L2-resident).  grid = (N/32, M/128).
// Epilogue: if scale_g: out = out*(scale_g[m]*scale_mul) + beta[m]
// ---------------------------------------------------------------------------
__global__ __launch_bounds__(256)
void gemm1x1_wmma_f32(const float* __restrict__ W,
                      const float* __restrict__ X,
                      float* __restrict__ Out,
                      int M, int K,
                      const float* __restrict__ scale_g,
                      const float* __restrict__ beta,
                      float scale_mul) {
  extern __shared__ float lds[];

  const int lane  = threadIdx.x & 31;
  const int wid   = threadIdx.x >> 5;
  const int nlane = lane & 15;
  const int kbase = (lane >> 4) << 1;        // K offset 0 / 2 per half-wave

  const int m0   = (blockIdx.y << 7) + (wid << 4);
  const int nblk = blockIdx.x << 5;          // 32 pixels per block
  const int b    = nblk >> 14;               // 32 | 16384 -> never straddles batch
  const int p0   = nblk & (HW - 1);

  // A fragment: W[m0 + nlane, kbase + k]  (K contiguous -> b64 loads)
  const float* __restrict__ Wt = W + (size_t)(m0 + nlane) * K + kbase;
  // X base for this block's 32-pixel column
  const float* __restrict__ Xb = X + (size_t)b * K * HW + p0;

  const int nchunks = K / KC;
  if (wid == 0) tdm_load_chunk(Xb, 0, lds, lane);

  v8f acc0 = {};
  v8f acc1 = {};

  for (int ch = 0; ch < nchunks; ++ch) {
    if (wid == 0) tdm_wait0();               // chunk ch landed in LDS
    __syncthreads();                         // all waves see it
    if (wid == 0 && (ch + 1) < nchunks) {
      // buffer (ch+1)&1 was last consumed in iteration ch-1, which every wave
      // finished before the barrier above -> safe to overwrite now.
      tdm_load_chunk(Xb + (size_t)(ch + 1) * KC * HW,
                     (unsigned)(((ch + 1) & 1) * (CHUNK_F * 4)), lds, lane);
    }
    const int k0 = ch * KC;
    __builtin_prefetch(Wt + k0 + KC, 0, 0);  // next A chunk (global_prefetch_b8)
    const float* __restrict__ Lb = lds + (ch & 1) * CHUNK_F + nlane;

#pragma unroll 4
    for (int kk = 0; kk < KC; kk += 4) {
      v2f a;
      a.x = Wt[k0 + kk];
      a.y = Wt[k0 + kk + 1];
      const int r = (kk + kbase) * LDS_ROW;
      v2f b0, b1;
      b0.x = Lb[r];
      b0.y = Lb[r + LDS_ROW];
      b1.x = Lb[r + 16];
      b1.y = Lb[r + LDS_ROW + 16];
      acc0 = __builtin_amdgcn_wmma_f32_16x16x4_f32(false, a, false, b0,
                                                   (short)0, acc0, false, false);
      acc1 = __builtin_amdgcn_wmma_f32_16x16x4_f32(false, a, false, b1,
                                                   (short)0, acc1, false, false);
    }
    __syncthreads();                         // done with buf ch&1 before reuse
  }

  // C/D layout: VGPR r -> M = m0 + r + 8*(lane>=16), N = nlane
  const int mhalf = (lane >> 4) << 3;
  float* __restrict__ Ct = Out + (size_t)b * M * HW + p0 + nlane;

  if (scale_g != nullptr) {
#pragma unroll
    for (int r = 0; r < 8; ++r) {
      const int m = m0 + mhalf + r;
      const float s = scale_g[m] * scale_mul;
      Ct[(size_t)m * HW]      = acc0[r] * s + beta[m];
      Ct[(size_t)m * HW + 16] = acc1[r] * s + beta[m];
    }
  } else {
#pragma unroll
    for (int r = 0; r < 8; ++r) {
      const int m = m0 + mhalf + r;
      Ct[(size_t)m * HW]      = acc0[r];
      Ct[(size_t)m * HW + 16] = acc1[r];
    }
  }
}

// ---------------------------------------------------------------------------
// Depthwise 5x5 conv, pad 2.  grid = (HW/256, B*768), block = 256.
// ---------------------------------------------------------------------------
__global__ __launch_bounds__(256)
void dwconv5x5(const float* __restrict__ in, const float* __restrict__ wdw,
               float* __restrict__ out) {
  const int bc = blockIdx.y;                       // b*768 + c
  const int c  = bc % 768;
  const int p  = (blockIdx.x << 8) + threadIdx.x;
  const int hy = p >> 7;
  const int wx = p & (HDIM - 1);

  const float* __restrict__ src = in + (size_t)bc * HW;
  const float* __restrict__ wt  = wdw + c * 25;

  float wreg[25];
#pragma unroll
  for (int i = 0; i < 25; ++i) wreg[i] = wt[i];

  float s = 0.0f;
#pragma unroll
  for (int dy = 0; dy < 5; ++dy) {
    const int y = hy + dy - 2;
    if ((unsigned)y < (unsigned)HDIM) {
      const float* row = src + y * HDIM;
#pragma unroll
      for (int dx = 0; dx < 5; ++dx) {
        const int x = wx + dx - 2;
        if ((unsigned)x < (unsigned)HDIM) s = fmaf(row[x], wreg[dy * 5 + dx], s);
      }
    }
  }
  out[(size_t)bc * HW + p] = s;
}

// ---------------------------------------------------------------------------
// Grouped pointwise conv: 96 groups, 8 in -> 8 out, in-place on agg.
// ---------------------------------------------------------------------------
__global__ __launch_bounds__(256)
void pwconv_grouped(float* __restrict__ agg, const float* __restrict__ wpw) {
  const int bg = blockIdx.y;
  const int g  = bg % 96;
  const int b  = bg / 96;
  const int p  = (blockIdx.x << 8) + threadIdx.x;

  float* __restrict__ base = agg + ((size_t)b * 768 + g * 8) * HW + p;
  const float* __restrict__ wt = wpw + g * 64;

  float vin[8];
#pragma unroll
  for (int i = 0; i < 8; ++i) vin[i] = base[(size_t)i * HW];

  float vout[8];
#pragma unroll
  for (int o = 0; o < 8; ++o) {
    float s = 0.0f;
#pragma unroll
    for (int i = 0; i < 8; ++i) s = fmaf(wt[o * 8 + i], vin[i], s);
    vout[o] = s;
  }
#pragma unroll
  for (int o = 0; o < 8; ++o) base[(size_t)o * HW] = vout[o];
}

// ---------------------------------------------------------------------------
// kv[bh] = sum_p relu(k[p,:8])^T (x) [v[p,:8] | 1]   (8x9 per (b,head))
// ---------------------------------------------------------------------------
__global__ __launch_bounds__(256)
void kv_reduce(const float* __restrict__ qkv, const float* __restrict__ agg,
               float* __restrict__ kvout) {
  const int bh = blockIdx.x;
  const int b  = bh >> 6;
  const int h  = bh & 63;
  const float* __restrict__ src = (h < 32) ? qkv : agg;
  const int hh = (h < 32) ? h : (h - 32);
  const float* __restrict__ base = src + ((size_t)b * 768 + hh * 24) * HW;

  __shared__ float skv[72];
  for (int i = threadIdx.x; i < 72; i += blockDim.x) skv[i] = 0.0f;
  __syncthreads();

  float acc[72];
#pragma unroll
  for (int i = 0; i < 72; ++i) acc[i] = 0.0f;

  for (int p = threadIdx.x; p < HW; p += blockDim.x) {
    float kd[8], ve[9];
#pragma unroll
    for (int d = 0; d < 8; ++d) kd[d] = fmaxf(base[(size_t)(8 + d) * HW + p], 0.0f);
#pragma unroll
    for (int e = 0; e < 8; ++e) ve[e] = base[(size_t)(16 + e) * HW + p];
    ve[8] = 1.0f;
#pragma unroll
    for (int d = 0; d < 8; ++d)
#pragma unroll
      for (int e = 0; e < 9; ++e) acc[d * 9 + e] = fmaf(kd[d], ve[e], acc[d * 9 + e]);
  }

#pragma unroll
  for (int i = 0; i < 72; ++i) atomicAdd(&skv[i], acc[i]);   // ds_add_f32
  __syncthreads();

  for (int i = threadIdx.x; i < 72; i += blockDim.x)
    kvout[(size_t)bh * 72 + i] = skv[i];
}

// ---------------------------------------------------------------------------
// out[p,:] = relu(q[p,:8]) @ kv(8x9); normalize by column 8 (+1e-15).
// ---------------------------------------------------------------------------
__global__ __launch_bounds__(256)
void attn_apply(const float* __restrict__ qkv, const float* __restrict__ agg,
                const float* __restrict__ kvin, float* __restrict__ attn) {
  const size_t idx = (size_t)blockIdx.x * 256 + threadIdx.x;
  const int p  = (int)(idx & (HW - 1));
  const int bh = (int)(idx >> 14);
  const int b  = bh >> 6;
  const int h  = bh & 63;

  const float* __restrict__ src = (h < 32) ? qkv : agg;
  const int hh = (h < 32) ? h : (h - 32);
  const float* __restrict__ base = src + ((size_t)b * 768 + hh * 24) * HW;
  const float* __restrict__ kv   = kvin + (size_t)bh * 72;

  float e[9];
#pragma unroll
  for (int j = 0; j < 9; ++j) e[j] = 0.0f;
#pragma unroll
  for (int d = 0; d < 8; ++d) {
    const float qd = fmaxf(base[(size_t)d * HW + p], 0.0f);
#pragma unroll
    for (int j = 0; j < 9; ++j) e[j] = fmaf(qd, kv[d * 9 + j], e[j]);
  }
  const float inv = 1.0f / (e[8] + 1e-15f);

  float* __restrict__ dst = attn + ((size_t)b * 512 + h * 8) * HW + p;
#pragma unroll
  for (int d = 0; d < 8; ++d) dst[(size_t)d * HW] = e[d] * inv;
}

// ---------------------------------------------------------------------------
extern "C" void kernel_launch(void* const* d_in, const int* in_sizes, int n_in,
                              void* d_out, int out_size, void* d_ws, size_t ws_size,
                              hipStream_t stream) {
  const float* x        = (const float*)d_in[0];  // (4,256,128,128)
  const float* w_qkv    = (const float*)d_in[1];  // (768,256)
  const float* w_dw     = (const float*)d_in[2];  // (768,25)
  const float* w_pw     = (const float*)d_in[3];  // (768,8)
  const float* w_proj   = (const float*)d_in[4];  // (256,512)
  const float* bn_gamma = (const float*)d_in[5];  // (256,)
  const float* bn_beta  = (const float*)d_in[6];  // (256,)
  float*       out      = (float*)d_out;          // (4,256,128,128)

  float* ws   = (float*)d_ws;
  float* qkv  = ws;                                // 768*65536 floats
  float* agg  = ws + 50331648ull;                  // 768*65536 floats
  float* attn = ws + 100663296ull;                 // 512*65536 floats
  float* kvb  = ws + 134217728ull;                 // 4*64*72 floats

  const int Npix = 4 * HW;                         // 65536

  // 1) qkv = Wqkv (768x256) * x               [WMMA f32 + TDM]
  {
    dim3 grid(Npix / 32, 768 / 128);
    gemm1x1_wmma_f32<<<grid, 256, LDS_BYTES, stream>>>(
        w_qkv, x, qkv, 768, 256, nullptr, nullptr, 1.0f);
  }
  // 2) depthwise 5x5 -> agg
  {
    dim3 grid(HW / 256, 4 * 768);
    dwconv5x5<<<grid, 256, 0, stream>>>(qkv, w_dw, agg);
  }
  // 3) grouped pointwise, in-place on agg (replay-safe: step 2 rewrites agg)
  {
    dim3 grid(HW / 256, 4 * 96);
    pwconv_grouped<<<grid, 256, 0, stream>>>(agg, w_pw);
  }
  // 4) kv reduction per (b, head)
  kv_reduce<<<4 * 64, 256, 0, stream>>>(qkv, agg, kvb);
  // 5) apply attention + normalize -> attn (B,512,HW)
  attn_apply<<<(4 * 64 * HW) / 256, 256, 0, stream>>>(qkv, agg, kvb, attn);
  // 6) proj = Wproj (256x512) * attn, fused BN  [WMMA f32 + TDM]
  {
    dim3 grid(Npix / 32, 256 / 128);
    gemm1x1_wmma_f32<<<grid, 256, LDS_BYTES, stream>>>(
        w_proj, attn, out, 256, 512, bn_gamma, bn_beta, 0.99999500003749969f);
  }
}